// LSTM_326417514597
// MI455X (gfx1250) — compile-verified
//
#include <hip/hip_runtime.h>

// MI455X / gfx1250 fused 4-layer LSTM (H=64) + FC head.
// 1 workgroup (8 wave32s) per 16-row batch tile runs the whole network;
// inter-layer activations ping-pong in LDS in 8-step time chunks.
// Recurrent matmul: full-precision V_WMMA_F32_16X16X4_F32.
// Cell state lives in LDS so the layer loop stays rolled -> <256 VGPRs,
// no spills, no s_set_vgpr_msb churn in the WMMA inner loop.

typedef __attribute__((ext_vector_type(2))) float v2f;
typedef __attribute__((ext_vector_type(8))) float v8f;

#define T_STEPS 512
#define BATCH   1024
#define HID     64
#define GATES   256      // 4*HID, PyTorch gate order i,f,g,o
#define TC      8        // timesteps per LDS chunk
#define NCHUNK  (T_STEPS / TC)
#define ROWS    16       // batch rows per workgroup (one WMMA M-tile)

static __device__ __forceinline__ v8f wmma_f32_k4(v2f a, v2f b, v8f c) {
  // D = A(16x4) * B(4x16) + C(16x16), all fp32.
  return __builtin_amdgcn_wmma_f32_16x16x4_f32(
      /*neg_a=*/false, a, /*neg_b=*/false, b,
      /*c_mod=*/(short)0, c, /*reuse_a=*/false, /*reuse_b=*/false);
}

static __device__ __forceinline__ float fast_sigmoid(float x) {
  return __builtin_amdgcn_rcpf(1.0f + __expf(-x));
}
static __device__ __forceinline__ float fast_tanh(float x) {
  return 2.0f * __builtin_amdgcn_rcpf(1.0f + __expf(-2.0f * x)) - 1.0f;
}

__global__ __launch_bounds__(256, 1)
void LSTM_326417514597_kernel(const float* __restrict__ x,          // [1024,512,4]
                              const float* __restrict__ w_ih0,      // [256,4]
                              const float* __restrict__ w_ih_rest,  // [3,256,64]
                              const float* __restrict__ w_hh,       // [4,256,64]
                              const float* __restrict__ b_ih,       // [4,256]
                              const float* __restrict__ b_hh,       // [4,256]
                              const float* __restrict__ w_fc,       // [5,64]
                              const float* __restrict__ b_fc,       // [5]
                              float* __restrict__ out)              // [1024,5]
{
  __shared__ float chA[TC * ROWS * HID];   // 32 KB  chunk activations (ping)
  __shared__ float chB[TC * ROWS * HID];   // 32 KB  chunk activations (pong)
  __shared__ float x0c[TC * ROWS * 4];     //  2 KB  staged raw input chunk
  __shared__ float gbuf[ROWS * GATES];     // 16 KB  pre-activation gates
  __shared__ float hsl[4 * ROWS * HID];    // 16 KB  per-layer hidden state
  __shared__ float csl[4 * ROWS * HID];    // 16 KB  per-layer cell state

  const int tid   = threadIdx.x;
  const int wg    = blockIdx.x;
  const int lane  = tid & 31;
  const int wave  = tid >> 5;
  const int row16 = lane & 15;          // M index of this lane's A fragment
  const int koff  = (lane >> 4) << 1;   // lanes 16..31 carry K=2,3 of each K-chunk
  const int nbase = wave * 32;          // each wave owns 2 N-tiles (32 gate cols)

  // h0 = 0, c0 = 0
  for (int q = tid; q < 4 * ROWS * HID; q += 256) { hsl[q] = 0.0f; csl[q] = 0.0f; }
  __syncthreads();

  for (int c = 0; c < NCHUNK; ++c) {
    // ---- stage x chunk: [TC][ROWS][4] ----
    for (int q = tid; q < TC * ROWS * 4; q += 256) {
      const int t = q >> 6, r = (q >> 2) & 15, i = q & 3;
      x0c[q] = x[((size_t)(wg * ROWS + r) * T_STEPS + (c * TC + t)) * 4 + i];
    }
    // Prefetch next chunk's x while we grind through 4 layers of compute.
    if (c + 1 < NCHUNK) {
      const int q = tid;           // 256 threads cover half the 512-float chunk;
      const int t = q >> 6, r = (q >> 2) & 15, i = (q & 3) * 2;  // stride-2 covers lines
      __builtin_prefetch(&x[((size_t)(wg * ROWS + r) * T_STEPS + ((c + 1) * TC + t)) * 4 + i], 0, 3);
    }
    __syncthreads();

    #pragma clang loop unroll(disable)
    for (int l = 0; l < 4; ++l) {
      // l0: out=chA | l1: in=chA out=chB | l2: in=chB out=chA | l3: in=chA out=chB
      float* inb  = (l & 1) ? chA : chB;  // valid for l >= 1
      float* outb = (l & 1) ? chB : chA;

      // ---- load this layer's B fragments (weights, K x N) into VGPRs ----
      // Assumed B frag layout for 16x16x4: v0 = B[k0+koff][n], v1 = B[k0+koff+1][n].
      // Features: k in [0,64) -> W_hh (h part), k >= 64 -> W_ih (x part).
      v2f bw0[32], bw1[32];
      const int n0 = nbase + row16;
      const int n1 = nbase + 16 + row16;
      #pragma unroll
      for (int kk = 0; kk < 16; ++kk) {
        bw0[kk] = *(const v2f*)&w_hh[((size_t)(l * GATES + n0)) * HID + 4 * kk + koff];
        bw1[kk] = *(const v2f*)&w_hh[((size_t)(l * GATES + n1)) * HID + 4 * kk + koff];
      }
      if (l == 0) {
        bw0[16] = *(const v2f*)&w_ih0[n0 * 4 + koff];
        bw1[16] = *(const v2f*)&w_ih0[n1 * 4 + koff];
        #pragma unroll
        for (int kk = 17; kk < 32; ++kk) { bw0[kk] = (v2f){0.f, 0.f}; bw1[kk] = (v2f){0.f, 0.f}; }
      } else {
        #pragma unroll
        for (int kk = 0; kk < 16; ++kk) {
          bw0[16 + kk] = *(const v2f*)&w_ih_rest[((size_t)((l - 1) * GATES + n0)) * HID + 4 * kk + koff];
          bw1[16 + kk] = *(const v2f*)&w_ih_rest[((size_t)((l - 1) * GATES + n1)) * HID + 4 * kk + koff];
        }
      }
      // Folded biases for the 4 cell elements this thread updates.
      float bsum[4][4];
      #pragma unroll
      for (int j = 0; j < 4; ++j) {
        const int hid = (tid * 4 + j) & 63;
        #pragma unroll
        for (int g = 0; g < 4; ++g)
          bsum[g][j] = b_ih[l * GATES + g * HID + hid] + b_hh[l * GATES + g * HID + hid];
      }

      // ---- TC recurrent steps ----
      #pragma clang loop unroll(disable)
      for (int ts = 0; ts < TC; ++ts) {
        v8f acc0 = {};
        v8f acc1 = {};
        // h part: K = 0..63
        const float* hrow = &hsl[(l * ROWS + row16) * HID];
        #pragma unroll
        for (int kk = 0; kk < 16; ++kk) {
          v2f a = *(const v2f*)&hrow[4 * kk + koff];
          acc0 = wmma_f32_k4(a, bw0[kk], acc0);
          acc1 = wmma_f32_k4(a, bw1[kk], acc1);
        }
        // x part
        if (l == 0) {  // in=4 -> exactly one K=4 chunk
          v2f a = *(const v2f*)&x0c[(ts * ROWS + row16) * 4 + koff];
          acc0 = wmma_f32_k4(a, bw0[16], acc0);
          acc1 = wmma_f32_k4(a, bw1[16], acc1);
        } else {       // in=64 from previous layer's LDS chunk
          const float* irow = &inb[(ts * ROWS + row16) * HID];
          #pragma unroll
          for (int kk = 0; kk < 16; ++kk) {
            v2f a = *(const v2f*)&irow[4 * kk + koff];
            acc0 = wmma_f32_k4(a, bw0[16 + kk], acc0);
            acc1 = wmma_f32_k4(a, bw1[16 + kk], acc1);
          }
        }
        // Scatter D tiles to the gate buffer (C/D layout: vgpr r, lanes<16 -> M=r).
        #pragma unroll
        for (int r = 0; r < 8; ++r) {
          const int m = ((lane >> 4) << 3) + r;
          gbuf[m * GATES + nbase + row16]      = acc0[r];
          gbuf[m * GATES + nbase + 16 + row16] = acc1[r];
        }
        __syncthreads();

        // ---- LSTM cell update (i,f,g,o); each thread owns 4 (row,hid) elems ----
        #pragma unroll
        for (int j = 0; j < 4; ++j) {
          const int e = tid * 4 + j, r = e >> 6, hid = e & 63;
          const float* gr = &gbuf[r * GATES + hid];
          const float gi = gr[0]        + bsum[0][j];
          const float gf = gr[HID]      + bsum[1][j];
          const float gg = gr[2 * HID]  + bsum[2][j];
          const float go = gr[3 * HID]  + bsum[3][j];
          const int   ci = (l * ROWS + r) * HID + hid;   // same thread owns csl[ci]
          const float cc = fast_sigmoid(gf) * csl[ci]
                         + fast_sigmoid(gi) * fast_tanh(gg);
          csl[ci] = cc;
          const float hh = fast_sigmoid(go) * fast_tanh(cc);
          hsl[ci]                           = hh;  // recurrent state (same index space)
          outb[(ts * ROWS + r) * HID + hid] = hh;  // feed next layer
        }
        __syncthreads();
      } // ts
    } // l
  } // chunks

  // ---- FC head on final h of layer 3 (t = T-1): out = h @ w_fc^T + b_fc ----
  if (tid < ROWS * 5) {
    const int r = tid / 5, j = tid % 5;
    float s = b_fc[j];
    #pragma unroll 8
    for (int k = 0; k < HID; ++k)
      s += hsl[(3 * ROWS + r) * HID + k] * w_fc[j * HID + k];
    out[(wg * ROWS + r) * 5 + j] = s;
  }
}

extern "C" void kernel_launch(void* const* d_in, const int* in_sizes, int n_in,
                              void* d_out, int out_size, void* d_ws, size_t ws_size,
                              hipStream_t stream) {
  (void)in_sizes; (void)n_in; (void)out_size; (void)d_ws; (void)ws_size;
  const float* x         = (const float*)d_in[0];
  const float* w_ih0     = (const float*)d_in[1];
  const float* w_ih_rest = (const float*)d_in[2];
  const float* w_hh      = (const float*)d_in[3];
  const float* b_ih      = (const float*)d_in[4];
  const float* b_hh      = (const float*)d_in[5];
  const float* w_fc      = (const float*)d_in[6];
  const float* b_fc      = (const float*)d_in[7];
  float* out = (float*)d_out;

  dim3 grid(BATCH / ROWS);  // 64 workgroups, one per 16-row batch tile
  dim3 block(256);          // 8 wave32s
  hipLaunchKernelGGL(LSTM_326417514597_kernel, grid, block, 0, stream,
                     x, w_ih0, w_ih_rest, w_hh, b_ih, b_hh, w_fc, b_fc, out);
}